// RobertSelfAttention_22454089024062
// MI455X (gfx1250) — compile-verified
//
#include <hip/hip_runtime.h>
#include <hip/hip_bf16.h>

typedef __attribute__((ext_vector_type(2))) float        v2f;
typedef __attribute__((ext_vector_type(8))) float        v8f;
typedef __attribute__((ext_vector_type(4))) unsigned int v4u;
typedef __attribute__((ext_vector_type(8))) int          v8i;
typedef __attribute__((ext_vector_type(4))) int          v4i;

#define NH    12
#define HS    64
#define SEQ   2048
#define HID   768
#define BATCH 2

#if defined(__gfx1250__) && __has_builtin(__builtin_amdgcn_tensor_load_to_lds)
#define HAVE_TDM 1
#else
#define HAVE_TDM 0
#endif

static __device__ __forceinline__ v8f wmma_f32_4(v2f a, v2f b, v8f c) {
  // D = A(16x4 f32) x B(4x16 f32) + C(16x16 f32)
  return __builtin_amdgcn_wmma_f32_16x16x4_f32(
      false, a, false, b, (short)0, c, false, false);
}

#if HAVE_TDM
// Issue a TDM load of a [16 rows x 64 f32] tile (row stride 64 elems) from
// global memory into LDS at byte offset lds_off.  D# fields per ISA 8.3/8.4.
static __device__ __forceinline__ void tdm_load_tile(unsigned lds_off,
                                                     const float* gptr) {
  unsigned long long ga = (unsigned long long)(uintptr_t)gptr;
  v4u g0 = {
      1u,                                   // count=1, user descriptor
      lds_off,                              // lds_addr (bytes)
      (unsigned)ga,                         // global_addr[31:0]
      ((unsigned)(ga >> 32) & 0x01FFFFFFu)  // global_addr[56:32]
          | 0x80000000u                     // type=2 ("image")
  };
  v8i g1 = {
      (int)(2u << 16),                 // data_size=2 (4 bytes)
      (int)((HS & 0xFFFFu) << 16),     // tensor_dim0 = 64  (bits 79:48)
      (int)((SEQ & 0xFFFFu) << 16),    // tensor_dim1 = 2048 (bits 111:80)
      (int)((HS & 0xFFFFu) << 16),     // tile_dim0 = 64    (bits 127:112)
      16,                              // tile_dim1 = 16, tile_dim2 = 0
      HS,                              // tensor_dim0_stride = 64 elems
      0, 0                             // tensor_dim1_stride = 0 (2-D)
  };
  v4i z = {0, 0, 0, 0};
#if __has_include(<hip/amd_detail/amd_gfx1250_TDM.h>)
  v8i z8 = {0, 0, 0, 0, 0, 0, 0, 0};
  __builtin_amdgcn_tensor_load_to_lds(g0, g1, z, z, z8, 0);   // clang-23 form
#else
  __builtin_amdgcn_tensor_load_to_lds(g0, g1, z, z, 0);       // ROCm 7.2 form
#endif
}
#endif

// ---------------------------------------------------------------------------
// Kernel 1: fused QKV projection, y = x @ W.T + b, scattered head-major.
// One wave per 64x64 output tile: 4x4 v8f accumulators (128 VGPRs),
// 16 WMMAs per 8 float2 loads -> 4x the math/byte of a 16x16 tile.
// ---------------------------------------------------------------------------
__global__ __launch_bounds__(32) void qkv_proj_kernel(
    const float* __restrict__ X,
    const float* __restrict__ Wq, const float* __restrict__ bq,
    const float* __restrict__ Wk, const float* __restrict__ bk,
    const float* __restrict__ Wv, const float* __restrict__ bv,
    float* __restrict__ Q, float* __restrict__ K, float* __restrict__ V) {
  const int lane = threadIdx.x;
  const int lo = lane & 15;
  const int hi = lane >> 4;
  const int m0 = blockIdx.x * 64;   // token tile over B*S = 4096
  const int n0 = blockIdx.y * 64;   // feature tile over H = 768

  const float* W;
  const float* bias;
  float* O;
  if (blockIdx.z == 0)      { W = Wq; bias = bq; O = Q; }
  else if (blockIdx.z == 1) { W = Wk; bias = bk; O = K; }
  else                      { W = Wv; bias = bv; O = V; }

  const float* ar[4];
  const float* br[4];
#pragma unroll
  for (int i = 0; i < 4; ++i) {
    ar[i] = X + (size_t)(m0 + 16 * i + lo) * HID + hi * 2;
    br[i] = W + (size_t)(n0 + 16 * i + lo) * HID + hi * 2;
  }

  v8f acc[4][4] = {};
  for (int k = 0; k < HID; k += 4) {
    v2f a[4], b[4];
#pragma unroll
    for (int i = 0; i < 4; ++i) a[i] = *(const v2f*)(ar[i] + k);
#pragma unroll
    for (int j = 0; j < 4; ++j) b[j] = *(const v2f*)(br[j] + k);
#pragma unroll
    for (int i = 0; i < 4; ++i)
#pragma unroll
      for (int j = 0; j < 4; ++j)
        acc[i][j] = wmma_f32_4(a[i], b[j], acc[i][j]);
  }

#pragma unroll
  for (int j = 0; j < 4; ++j) {
    const int col = n0 + 16 * j + lo;
    const float bn = bias[col];
    const int h = col >> 6;
    const int d = col & 63;
#pragma unroll
    for (int i = 0; i < 4; ++i) {
#pragma unroll
      for (int r = 0; r < 8; ++r) {
        const int row = m0 + 16 * i + r + hi * 8;   // token in [0, 4096)
        const int bi = row >> 11;
        const int s  = row & (SEQ - 1);
        O[((size_t)(bi * NH + h) * SEQ + s) * HS + d] = acc[i][j][r] + bn;
      }
    }
  }
}

// ---------------------------------------------------------------------------
// Kernel 2: flash-style attention.  4 waves / block, 64 queries / block.
// K/V 16x64 tiles are double-buffered in LDS; wave 0 issues TDM loads for
// tile kt+1 before compute of tile kt and waits TENSORcnt after, overlapping
// the DMA with 32 WMMAs + online softmax.
// ---------------------------------------------------------------------------
__global__ __launch_bounds__(128) void attention_kernel(
    const float* __restrict__ Q, const float* __restrict__ K,
    const float* __restrict__ V, const int* __restrict__ mask,
    float* __restrict__ out) {
  __shared__ float k_lds[2][16][HS];
  __shared__ float v_lds[2][16][HS];
  __shared__ float p_lds[4][16][16];

  const float NEG_INF = -__builtin_inff();
  const int tid  = threadIdx.x;
  const int wave = tid >> 5;
  const int lane = tid & 31;
  const int lo = lane & 15;
  const int hi = lane >> 4;
  const int q0 = blockIdx.x * 64 + wave * 16;
  const int h  = blockIdx.y;
  const int b  = blockIdx.z;

  const size_t head_off = (size_t)(b * NH + h) * SEQ * HS;
  const float* Qh = Q + head_off;
  const float* Kh = K + head_off;
  const float* Vh = V + head_off;

  // Preload this wave's 16 Q A-fragments (K-dim 64 in steps of 4).
  v2f aq[16];
  {
    const float* qrow = Qh + (size_t)(q0 + lo) * HS;
#pragma unroll
    for (int t = 0; t < 16; ++t)
      aq[t] = *(const v2f*)(qrow + 4 * t + hi * 2);
  }

  float mstat[8], lstat[8];
#pragma unroll
  for (int r = 0; r < 8; ++r) { mstat[r] = NEG_INF; lstat[r] = 0.0f; }
  v8f o[4] = {};

  const int NT = SEQ / 16;

  // ---- prologue: stage tile 0 into buffer 0 ----
#if HAVE_TDM
  if (wave == 0) {
    tdm_load_tile((unsigned)(uintptr_t)&k_lds[0][0][0], Kh);
    tdm_load_tile((unsigned)(uintptr_t)&v_lds[0][0][0], Vh);
    __builtin_amdgcn_s_wait_tensorcnt(0);
  }
#else
  for (int i = tid; i < 16 * HS / 2; i += 128) {
    ((v2f*)&k_lds[0][0][0])[i] = ((const v2f*)Kh)[i];
    ((v2f*)&v_lds[0][0][0])[i] = ((const v2f*)Vh)[i];
  }
#endif
  __syncthreads();

  int cur = 0;
  for (int kt = 0; kt < NT; ++kt) {
    // ---- prefetch next K/V tile into the other buffer (async DMA) ----
#if HAVE_TDM
    if (wave == 0 && kt + 1 < NT) {
      const size_t nxt = (size_t)(kt + 1) * 16 * HS;
      tdm_load_tile((unsigned)(uintptr_t)&k_lds[cur ^ 1][0][0], Kh + nxt);
      tdm_load_tile((unsigned)(uintptr_t)&v_lds[cur ^ 1][0][0], Vh + nxt);
    }
#endif

    // ---- scores = Q · K^T for this 16x16 tile (K from LDS) ----
    v8f sacc = {};
#pragma unroll
    for (int t = 0; t < 16; ++t) {
      v2f bk = *(const v2f*)(&k_lds[cur][lo][4 * t + hi * 2]);
      sacc = wmma_f32_4(aq[t], bk, sacc);
    }

    // ---- scale + mask ----
    const int keycol = kt * 16 + lo;
    const bool keep = mask[b * SEQ + keycol] != 0;
    float sv[8];
#pragma unroll
    for (int r = 0; r < 8; ++r)
      sv[r] = keep ? sacc[r] * 0.125f : NEG_INF;   // 1/sqrt(64)

    // ---- online softmax per row (row = one 16-lane half of the wave) ----
#pragma unroll
    for (int r = 0; r < 8; ++r) {
      float rmax = sv[r];
      rmax = fmaxf(rmax, __shfl_xor(rmax, 1, 32));
      rmax = fmaxf(rmax, __shfl_xor(rmax, 2, 32));
      rmax = fmaxf(rmax, __shfl_xor(rmax, 4, 32));
      rmax = fmaxf(rmax, __shfl_xor(rmax, 8, 32));
      const float mnew = fmaxf(mstat[r], rmax);
      const float factor = (mnew == NEG_INF) ? 1.0f : __expf(mstat[r] - mnew);
      const float p = (sv[r] == NEG_INF) ? 0.0f : __expf(sv[r] - mnew);
      float rsum = p;
      rsum += __shfl_xor(rsum, 1, 32);
      rsum += __shfl_xor(rsum, 2, 32);
      rsum += __shfl_xor(rsum, 4, 32);
      rsum += __shfl_xor(rsum, 8, 32);
      lstat[r] = lstat[r] * factor + rsum;
      mstat[r] = mnew;
#pragma unroll
      for (int c = 0; c < 4; ++c) o[c][r] *= factor;
      p_lds[wave][r + hi * 8][lo] = p;      // C-layout -> LDS
    }
    __syncthreads();   // p visible; uniform across all 4 waves

    // ---- O += P · V  (P from LDS in A-layout, V from LDS) ----
#pragma unroll
    for (int t = 0; t < 4; ++t) {
      v2f ap = *(const v2f*)(&p_lds[wave][lo][4 * t + hi * 2]);
      const int k0 = 4 * t + hi * 2;
#pragma unroll
      for (int c = 0; c < 4; ++c) {
        v2f bv = { v_lds[cur][k0][c * 16 + lo], v_lds[cur][k0 + 1][c * 16 + lo] };
        o[c] = wmma_f32_4(ap, bv, o[c]);
      }
    }

#if HAVE_TDM
    if (wave == 0 && kt + 1 < NT) __builtin_amdgcn_s_wait_tensorcnt(0);
#else
    __syncthreads();
    if (kt + 1 < NT) {
      const size_t nxt = (size_t)(kt + 1) * 16 * HS;
      for (int i = tid; i < 16 * HS / 2; i += 128) {
        ((v2f*)&k_lds[cur ^ 1][0][0])[i] = ((const v2f*)(Kh + nxt))[i];
        ((v2f*)&v_lds[cur ^ 1][0][0])[i] = ((const v2f*)(Vh + nxt))[i];
      }
    }
#endif
    __syncthreads();   // tile kt+1 staged & everyone done with buffer `cur`
    cur ^= 1;
  }

  // ---- normalize and write out [B, S, H] ----
  float inv[8];
#pragma unroll
  for (int r = 0; r < 8; ++r) inv[r] = (lstat[r] > 0.0f) ? 1.0f / lstat[r] : 0.0f;
#pragma unroll
  for (int r = 0; r < 8; ++r) {
    const int srow = q0 + r + hi * 8;
    float* orow = out + (size_t)(b * SEQ + srow) * HID + h * HS;
#pragma unroll
    for (int c = 0; c < 4; ++c)
      orow[c * 16 + lo] = o[c][r] * inv[r];
  }
}

extern "C" void kernel_launch(void* const* d_in, const int* in_sizes, int n_in,
                              void* d_out, int out_size, void* d_ws, size_t ws_size,
                              hipStream_t stream) {
  const float* X    = (const float*)d_in[0];
  const int*   mask = (const int*)d_in[1];
  const float* Wq   = (const float*)d_in[2];
  const float* bq   = (const float*)d_in[3];
  const float* Wk   = (const float*)d_in[4];
  const float* bk   = (const float*)d_in[5];
  const float* Wv   = (const float*)d_in[6];
  const float* bv   = (const float*)d_in[7];
  float* out = (float*)d_out;

  const size_t per = (size_t)BATCH * SEQ * HID;
  float* Qb = (float*)d_ws;
  float* Kb = Qb + per;
  float* Vb = Kb + per;

  qkv_proj_kernel<<<dim3((BATCH * SEQ) / 64, HID / 64, 3), 32, 0, stream>>>(
      X, Wq, bq, Wk, bk, Wv, bv, Qb, Kb, Vb);
  attention_kernel<<<dim3(SEQ / 64, NH, BATCH), 128, 0, stream>>>(
      Qb, Kb, Vb, mask, out);
}